// S4Layer_54417235640984
// MI455X (gfx1250) — compile-verified
//
#include <hip/hip_runtime.h>

typedef float v2f __attribute__((ext_vector_type(2)));
typedef float v8f __attribute__((ext_vector_type(8)));

#define D_MODEL 512
#define D_STATE 32
#define BATCH 8
#define SEQ 4096
#define ROWS (BATCH * SEQ) /* 32768 */

/* workspace layout (floats) */
#define WS_ABAR 0                         /* 32*32      */
#define WS_BBAR (WS_ABAR + 1024)          /* 32*512     */
#define WS_UB (WS_BBAR + 16384)           /* ROWS*32    */
#define WS_STATES (WS_UB + ROWS * 32)     /* ROWS*32    */

__device__ __forceinline__ v8f wmma4(v2f a, v2f b, v8f c) {
  /* D = A(16x4,f32) x B(4x16,f32) + C(16x16,f32) -> v_wmma_f32_16x16x4_f32 */
  return __builtin_amdgcn_wmma_f32_16x16x4_f32(false, a, false, b, (short)0, c,
                                               false, false);
}

/* ---------------- prep: dt, inv(I-0.5dtA), A_bar, B_bar ---------------- */
__global__ void __launch_bounds__(256) prep_kernel(
    const float* __restrict__ A, const float* __restrict__ B,
    const float* __restrict__ log_dt, float* __restrict__ ws) {
  __shared__ float aug[32][64];
  __shared__ float Ah[32][32];
  __shared__ float inv[32][32];
  __shared__ float fac[32];
  __shared__ float dt_sh;
  const int tid = threadIdx.x;

  if (tid == 0) {
    float s = 0.f;
    for (int i = 0; i < D_MODEL; ++i) {
      float v = expf(log_dt[i]);
      v = fminf(fmaxf(v, 1e-4f), 1.0f);
      s += v;
    }
    dt_sh = s / (float)D_MODEL;
  }
  __syncthreads();
  const float dt = dt_sh;

  for (int idx = tid; idx < 32 * 32; idx += blockDim.x) {
    const int i = idx >> 5, j = idx & 31;
    const float a = A[idx];
    const float eye = (i == j) ? 1.f : 0.f;
    aug[i][j] = eye - 0.5f * dt * a;
    aug[i][j + 32] = eye;
    Ah[i][j] = eye + 0.5f * dt * a;
  }
  __syncthreads();

  /* Gauss-Jordan, no pivoting (matrix is strongly diagonally dominated here) */
  for (int k = 0; k < 32; ++k) {
    const float p = aug[k][k];
    __syncthreads();
    for (int j = tid; j < 64; j += blockDim.x) aug[k][j] = aug[k][j] / p;
    __syncthreads();
    if (tid < 32) fac[tid] = aug[tid][k];
    __syncthreads();
    for (int idx = tid; idx < 32 * 64; idx += blockDim.x) {
      const int i = idx >> 6, j = idx & 63;
      if (i != k) aug[i][j] -= fac[i] * aug[k][j];
    }
    __syncthreads();
  }
  for (int idx = tid; idx < 32 * 32; idx += blockDim.x) {
    const int i = idx >> 5, j = idx & 31;
    inv[i][j] = aug[i][j + 32];
  }
  __syncthreads();

  float* Abar = ws + WS_ABAR;
  for (int idx = tid; idx < 32 * 32; idx += blockDim.x) {
    const int i = idx >> 5, j = idx & 31;
    float s = 0.f;
    for (int k = 0; k < 32; ++k) s += Ah[i][k] * inv[k][j];
    Abar[idx] = s;
  }
  float* Bbar = ws + WS_BBAR;
  for (int idx = tid; idx < 32 * D_MODEL; idx += blockDim.x) {
    const int n = idx >> 9, d = idx & 511;
    float s = 0.f;
    for (int k = 0; k < 32; ++k) s += inv[n][k] * B[k * D_MODEL + d];
    Bbar[idx] = dt * s;
  }
}

/* ---------------- uB = x @ B_bar^T   (M=32768, K=512, N=32) ------------ */
__global__ void __launch_bounds__(256) ub_kernel(const float* __restrict__ x,
                                                 const float* __restrict__ Bbar,
                                                 float* __restrict__ uB) {
  const int wave = threadIdx.x >> 5;
  const int lane = threadIdx.x & 31;
  const int r = lane & 15, h = lane >> 4;
  const int mtile = blockIdx.x * 8 + wave; /* 0..2047 */

  const float* xrow = x + (size_t)(mtile * 16 + r) * D_MODEL;
  const float* b0row = Bbar + (size_t)r * D_MODEL;        /* N = 0..15  */
  const float* b1row = Bbar + (size_t)(16 + r) * D_MODEL; /* N = 16..31 */

  v8f acc0 = {0.f, 0.f, 0.f, 0.f, 0.f, 0.f, 0.f, 0.f};
  v8f acc1 = {0.f, 0.f, 0.f, 0.f, 0.f, 0.f, 0.f, 0.f};

  for (int k0 = 0; k0 < D_MODEL; k0 += 4) {
    const int k = k0 + 2 * h;
    const v2f a = *(const v2f*)(xrow + k);
    const v2f b0 = *(const v2f*)(b0row + k);
    const v2f b1 = *(const v2f*)(b1row + k);
    acc0 = wmma4(a, b0, acc0);
    acc1 = wmma4(a, b1, acc1);
  }

  const int mbase = mtile * 16 + 8 * h;
#pragma unroll
  for (int v = 0; v < 8; ++v) {
    const int m = mbase + v;
    uB[(size_t)m * 32 + r] = acc0[v];
    uB[(size_t)m * 32 + 16 + r] = acc1[v];
  }
}

/* ------------- sequential tanh scan: 1 wave per batch ------------------ */
__global__ void __launch_bounds__(32) scan_kernel(const float* __restrict__ ws,
                                                  float* __restrict__ states,
                                                  float* __restrict__ fs_out) {
  const int b = blockIdx.x;
  const int lane = threadIdx.x; /* state element n */
  const float* Abar = ws + WS_ABAR;
  const float* uB = ws + WS_UB + (size_t)b * SEQ * 32;
  float* st_out = states + (size_t)b * SEQ * 32;

  float a[32];
#pragma unroll
  for (int m = 0; m < 32; ++m) a[m] = Abar[lane * 32 + m];

  float st = 0.f;
  for (int s = 0; s < SEQ; ++s) {
    float acc = uB[s * 32 + lane];
    const int stb = __float_as_int(st);
#pragma unroll
    for (int m = 0; m < 32; ++m) {
      /* broadcast state[m] to all lanes via v_readlane -> SGPR operand FMA */
      acc = fmaf(a[m], __int_as_float(__builtin_amdgcn_readlane(stb, m)), acc);
    }
    st = tanhf(acc);
    st_out[s * 32 + lane] = st;
  }
  fs_out[b * 32 + lane] = st;
}

/* --------- y1 = states @ C^T + x*D  (M=32768, K=32, N=512) ------------- */
__global__ void __launch_bounds__(256) cproj_kernel(
    const float* __restrict__ states, const float* __restrict__ C,
    const float* __restrict__ x, const float* __restrict__ Dv,
    float* __restrict__ y1) {
  const int wave = threadIdx.x >> 5;
  const int lane = threadIdx.x & 31;
  const int r = lane & 15, h = lane >> 4;
  const int mtile = blockIdx.x * 8 + wave;

  v2f afrag[8];
  const float* srow = states + (size_t)(mtile * 16 + r) * 32;
#pragma unroll
  for (int kk = 0; kk < 8; ++kk)
    afrag[kk] = *(const v2f*)(srow + kk * 4 + 2 * h);

  const int mbase = mtile * 16 + 8 * h;
  for (int dtile = 0; dtile < 32; ++dtile) {
    v8f acc = {0.f, 0.f, 0.f, 0.f, 0.f, 0.f, 0.f, 0.f};
    const float* crow = C + (size_t)(dtile * 16 + r) * 32;
#pragma unroll
    for (int kk = 0; kk < 8; ++kk) {
      const v2f b = *(const v2f*)(crow + kk * 4 + 2 * h);
      acc = wmma4(afrag[kk], b, acc);
    }
    const int d = dtile * 16 + r;
    const float Dd = Dv[d];
#pragma unroll
    for (int v = 0; v < 8; ++v) {
      const int m = mbase + v;
      y1[(size_t)m * D_MODEL + d] = acc[v] + x[(size_t)m * D_MODEL + d] * Dd;
    }
  }
}

/* -------- y = y1 @ W_out^T + b_out, in place on d_out ------------------ */
__global__ void __launch_bounds__(256) wout_kernel(
    const float* __restrict__ W, const float* __restrict__ bout,
    float* __restrict__ y) {
  __shared__ float lds[16 * D_MODEL]; /* 32 KB tile of y1 */
  const int tid = threadIdx.x;
  const int mtile = blockIdx.x;

  /* stage the 16 rows this block owns; after barrier we may overwrite them */
  const float4* src4 = (const float4*)(y + (size_t)mtile * 16 * D_MODEL);
  float4* dst4 = (float4*)lds;
#pragma unroll
  for (int i = 0; i < 8; ++i) dst4[tid + i * 256] = src4[tid + i * 256];
  __syncthreads();

  const int wave = tid >> 5, lane = tid & 31;
  const int r = lane & 15, h = lane >> 4;
  const int mbase = mtile * 16 + 8 * h;
  const float* arow = lds + r * D_MODEL;

  for (int t = 0; t < 4; ++t) {
    const int ntile = wave * 4 + t;
    const float* wrow = W + (size_t)(ntile * 16 + r) * D_MODEL;
    if (t < 3)
      __builtin_prefetch(W + (size_t)((ntile + 1) * 16 + r) * D_MODEL, 0, 1);

    v8f acc = {0.f, 0.f, 0.f, 0.f, 0.f, 0.f, 0.f, 0.f};
    for (int k0 = 0; k0 < D_MODEL; k0 += 4) {
      const int k = k0 + 2 * h;
      const v2f a = *(const v2f*)(arow + k); /* ds_load feeding WMMA */
      const v2f b = *(const v2f*)(wrow + k);
      acc = wmma4(a, b, acc);
    }
    const int n = ntile * 16 + r;
    const float bo = bout[n];
#pragma unroll
    for (int v = 0; v < 8; ++v)
      y[(size_t)(mbase + v) * D_MODEL + n] = acc[v] + bo;
  }
}

extern "C" void kernel_launch(void* const* d_in, const int* in_sizes, int n_in,
                              void* d_out, int out_size, void* d_ws,
                              size_t ws_size, hipStream_t stream) {
  const float* x = (const float*)d_in[0];
  const float* A = (const float*)d_in[1];
  const float* B = (const float*)d_in[2];
  const float* C = (const float*)d_in[3];
  const float* Dv = (const float*)d_in[4];
  const float* log_dt = (const float*)d_in[5];
  const float* W = (const float*)d_in[6];
  const float* bout = (const float*)d_in[7];

  float* y = (float*)d_out;                      /* (8,4096,512) */
  float* fs = y + (size_t)ROWS * D_MODEL;        /* (8,32) final state */

  float* ws = (float*)d_ws;
  float* Bbar = ws + WS_BBAR;
  float* uB = ws + WS_UB;
  float* states = ws + WS_STATES;

  prep_kernel<<<1, 256, 0, stream>>>(A, B, log_dt, ws);
  ub_kernel<<<256, 256, 0, stream>>>(x, Bbar, uB);
  scan_kernel<<<BATCH, 32, 0, stream>>>(ws, states, fs);
  cproj_kernel<<<256, 256, 0, stream>>>(states, C, x, Dv, y);
  wout_kernel<<<2048, 256, 0, stream>>>(W, bout, y);
}